// VQLayer_21586505630024
// MI455X (gfx1250) — compile-verified
//
#include <hip/hip_runtime.h>
#include <math.h>

// Problem constants (match reference)
#define NROWS 32768
#define KPROT 1024
#define DDIM  256
#define MTILE 16        // rows per workgroup
#define EPSV  1e-6f

typedef float v2f __attribute__((ext_vector_type(2)));
typedef float v8f __attribute__((ext_vector_type(8)));

// ---------------------------------------------------------------------------
// Kernel 0: prototype squared norms + zero the column accumulators.
// 128 blocks x 256 threads; one wave32 per prototype row.
// ---------------------------------------------------------------------------
__global__ void vq_init_kernel(const float* __restrict__ P,
                               float* __restrict__ sums,   // [2*KPROT] (Ssoft, Slogp)
                               float* __restrict__ pn)     // [KPROT]
{
    int tid = blockIdx.x * blockDim.x + threadIdx.x;
    if (tid < 2 * KPROT) sums[tid] = 0.0f;

    int w    = threadIdx.x >> 5;
    int lane = threadIdx.x & 31;
    int row  = blockIdx.x * 8 + w;          // 128*8 = 1024 rows
    const float* pr = P + (size_t)row * DDIM;
    float part = 0.0f;
#pragma unroll
    for (int i = 0; i < 8; ++i) {
        float v = pr[lane + 32 * i];
        part += v * v;
    }
#pragma unroll
    for (int off = 16; off; off >>= 1) part += __shfl_xor(part, off, 32);
    if (lane == 0) pn[row] = part;
}

// ---------------------------------------------------------------------------
// Kernel 1: fused distances (WMMA f32) + gumbel + log-softmax + argmax gather
//           + column-sum atomics.  grid = NROWS/16, block = 256 (8 waves).
// Dynamic LDS layout (floats):
//   latA  [16*256]   staged latents tile
//   lg    [16*1024]  logits tile
//   xnS   [16]       row squared norms
//   lseS  [16]       row log-sum-exp
//   red   [256]      reduction scratch
// ---------------------------------------------------------------------------
__global__ void vq_main_kernel(const float* __restrict__ lat,
                               const float* __restrict__ P,
                               const float* __restrict__ gmb,
                               const float* __restrict__ pn,
                               float* __restrict__ outQ,     // [N, D]
                               float* __restrict__ Ssoft,    // [K]
                               float* __restrict__ Slogp)    // [K]
{
    extern __shared__ float smem[];
    float* latA = smem;                    // 4096
    float* lg   = latA + MTILE * DDIM;     // 16384
    float* xnS  = lg + MTILE * KPROT;      // 16
    float* lseS = xnS + MTILE;             // 16
    float* red  = lseS + MTILE;            // 256

    const int tid  = threadIdx.x;
    const int w    = tid >> 5;
    const int lane = tid & 31;
    const int half = lane >> 4;            // 0: lanes 0-15, 1: lanes 16-31
    const int nIdx = lane & 15;
    const int rowBase = blockIdx.x * MTILE;

    // ---- stage latents tile (coalesced) ----
    {
        const float* src = lat + (size_t)rowBase * DDIM;
#pragma unroll
        for (int i = 0; i < MTILE * DDIM / 256; ++i) {
            int idx = tid + 256 * i;
            latA[idx] = src[idx];
        }
    }
    __syncthreads();

    // ---- row squared norms ----
    {
        int r = tid >> 4, j = tid & 15;
        float p = 0.0f;
#pragma unroll
        for (int i = 0; i < 16; ++i) p += latA[r * DDIM + j * 16 + i] * latA[r * DDIM + j * 16 + i];
        red[tid] = p;
    }
    __syncthreads();
    if (tid < MTILE) {
        float s = 0.0f;
#pragma unroll
        for (int i = 0; i < 16; ++i) s += red[tid * 16 + i];
        xnS[tid] = s;
    }
    __syncthreads();

    // ---- WMMA GEMM: 16 rows x 128 cols per wave, 8 tiles of 16x16 ----
    // f32 16x16x4 A layout: lane(0-15)=row M, VGPR0=K(half*2), VGPR1=K(half*2+1)
    const int kOff = half * 2;
    const int rowA = nIdx;
    for (int t = 0; t < 8; ++t) {
        const int ct = w * 128 + t * 16;
        v8f acc = {0.f, 0.f, 0.f, 0.f, 0.f, 0.f, 0.f, 0.f};
        const float* bCol = P + (size_t)(ct + nIdx) * DDIM;   // B[k][n] = P[col n][k]
#pragma unroll 4
        for (int kk = 0; kk < DDIM; kk += 4) {
            v2f a, b;
            a.x = latA[rowA * DDIM + kk + kOff];
            a.y = latA[rowA * DDIM + kk + kOff + 1];
            b.x = bCol[kk + kOff];
            b.y = bCol[kk + kOff + 1];
            acc = __builtin_amdgcn_wmma_f32_16x16x4_f32(
                false, a, false, b, (short)0, acc, false, false);
        }
        // epilogue: logits = 2*dot - ||x||^2 - ||p||^2 + gumbel
        const float pncol = pn[ct + nIdx];
#pragma unroll
        for (int r = 0; r < 8; ++r) {
            int row = r + half * 8;
            float g = gmb[(size_t)(rowBase + row) * KPROT + ct + nIdx];
            lg[row * KPROT + ct + nIdx] = 2.0f * acc[r] - xnS[row] - pncol + g;
        }
    }
    __syncthreads();

    // ---- per-row softmax stats, argmax, prototype gather (2 rows per wave) ----
#pragma unroll
    for (int rr = 0; rr < 2; ++rr) {
        const int row = w * 2 + rr;
        const float* lrow = lg + row * KPROT;
        float m = -INFINITY; int mi = 0;
        for (int c = lane; c < KPROT; c += 32) {
            float v = lrow[c];
            if (v > m) { m = v; mi = c; }
        }
#pragma unroll
        for (int off = 16; off; off >>= 1) {
            float om = __shfl_xor(m, off, 32);
            int   oi = __shfl_xor(mi, off, 32);
            if (om > m || (om == m && oi < mi)) { m = om; mi = oi; }
        }
        float s = 0.0f;
        for (int c = lane; c < KPROT; c += 32) s += __expf(lrow[c] - m);
#pragma unroll
        for (int off = 16; off; off >>= 1) s += __shfl_xor(s, off, 32);
        const float lse = m + __logf(s);
        if (lane == 0) lseS[row] = lse;

        // quantized[n] = prototypes[argmax]
        const float* src = P + (size_t)mi * DDIM;
        float* dst = outQ + (size_t)(rowBase + row) * DDIM;
#pragma unroll
        for (int i = 0; i < 8; ++i) dst[lane + 32 * i] = src[lane + 32 * i];
    }
    __syncthreads();

    // ---- column partial sums over the 16 rows, atomically accumulated ----
    for (int c = tid; c < KPROT; c += 256) {
        float ss = 0.0f, sl = 0.0f;
#pragma unroll
        for (int row = 0; row < MTILE; ++row) {
            float v = lg[row * KPROT + c] - lseS[row];   // logprob
            sl += v;
            ss += __expf(v);                              // y_soft
        }
        atomicAdd(&Ssoft[c], ss);
        atomicAdd(&Slogp[c], sl);
    }
}

// ---------------------------------------------------------------------------
// Kernel 2: final scalar loss.
// vq_loss = 1.001 * sum_k prior*log(prior) - sum_k prior * Slogp[k]/N
// with prior_k = Ssoft[k]/N + EPS
// ---------------------------------------------------------------------------
__global__ void vq_loss_kernel(const float* __restrict__ Ssoft,
                               const float* __restrict__ Slogp,
                               float* __restrict__ lossOut)
{
    __shared__ float red[256];
    const float invN = 1.0f / (float)NROWS;
    float part = 0.0f;
    for (int c = threadIdx.x; c < KPROT; c += 256) {
        float prior = Ssoft[c] * invN + EPSV;
        float lp = __logf(prior);
        part += prior * lp * 1.001f - prior * (Slogp[c] * invN);
    }
    red[threadIdx.x] = part;
    __syncthreads();
    for (int off = 128; off; off >>= 1) {
        if (threadIdx.x < off) red[threadIdx.x] += red[threadIdx.x + off];
        __syncthreads();
    }
    if (threadIdx.x == 0) *lossOut = red[0];
}

// ---------------------------------------------------------------------------
extern "C" void kernel_launch(void* const* d_in, const int* in_sizes, int n_in,
                              void* d_out, int out_size, void* d_ws, size_t ws_size,
                              hipStream_t stream) {
    const float* lat = (const float*)d_in[0];   // [N, D]
    const float* P   = (const float*)d_in[1];   // [K, D]
    const float* gmb = (const float*)d_in[2];   // [N, K]

    float* outQ    = (float*)d_out;                       // [N, D]
    float* lossOut = (float*)d_out + (size_t)NROWS * DDIM;

    float* ws    = (float*)d_ws;
    float* Ssoft = ws;              // [K]
    float* Slogp = ws + KPROT;      // [K]
    float* pn    = ws + 2 * KPROT;  // [K]

    vq_init_kernel<<<KPROT / 8, 256, 0, stream>>>(P, ws, pn);

    const size_t lds_bytes = (MTILE * DDIM + MTILE * KPROT + MTILE + MTILE + 256) * sizeof(float);
    vq_main_kernel<<<NROWS / MTILE, 256, lds_bytes, stream>>>(
        lat, P, gmb, pn, outQ, Ssoft, Slogp);

    vq_loss_kernel<<<1, 256, 0, stream>>>(Ssoft, Slogp, lossOut);
}